// GaussianMixtureModel_38311108280524
// MI455X (gfx1250) — compile-verified
//
#include <hip/hip_runtime.h>
#include <hip/hip_bf16.h>
#include <cmath>

// WMMA vector types (wave32, gfx1250)
typedef __attribute__((ext_vector_type(2))) float v2f;   // 16x4 f32 A / 4x16 f32 B: 2 VGPRs/lane
typedef __attribute__((ext_vector_type(8))) float v8f;   // 16x16 f32 C/D: 8 VGPRs/lane

#define NPTS 8192
#define NCOMP 64
#define ISPLIT 16
#define ISLICE (NPTS / ISPLIT)        // 512 floats staged in LDS per block (2 KB)
#define ROWS_PER_BLOCK 128            // 8 waves x 16 rows
#define LOG2E 1.4426950408889634f

// ---------------------------------------------------------------------------
// Kernel 1: softmax over 64 logits, per-component constants, zero accumulators
// ---------------------------------------------------------------------------
__global__ void gmm_prep_kernel(const float* __restrict__ wlog,
                                const float* __restrict__ lvar,
                                float* __restrict__ coef,
                                float* __restrict__ e2k,
                                float* __restrict__ S,
                                float* __restrict__ out)
{
    __shared__ float red[NCOMP];
    const int k = threadIdx.x;
    const float l = wlog[k];
    red[k] = l; __syncthreads();
    for (int s = NCOMP / 2; s > 0; s >>= 1) {
        if (k < s) red[k] = fmaxf(red[k], red[k + s]);
        __syncthreads();
    }
    const float m = red[0]; __syncthreads();
    const float e = __builtin_amdgcn_exp2f((l - m) * LOG2E);
    red[k] = e; __syncthreads();
    for (int s = NCOMP / 2; s > 0; s >>= 1) {
        if (k < s) red[k] += red[k + s];
        __syncthreads();
    }
    const float w   = e / red[0];                                  // softmax weight
    const float var = __builtin_amdgcn_exp2f(lvar[k] * LOG2E);     // exp(log_var)
    coef[k] = w / sqrtf(6.2831853071795864f * var);                // w / sqrt(2*pi*var)
    e2k[k]  = (-0.5f * LOG2E) / var;                               // exp2 coefficient
    for (int i = k; i < NPTS; i += NCOMP) S[i] = 0.0f;             // zero KDE row sums
    if (k == 0) out[0] = 0.0f;                                     // zero final scalar
}

// ---------------------------------------------------------------------------
// Kernel 2: pairwise KDE row sums. 16 rows/wave; lane (l&15)=row, (l>>4)=K pair
// of the 16x4 f32 A-matrix. LDS slice is pre-scaled by k=sqrt(-c2) so the
// per-element arg is -((k*xj)-(k*xi))^2: sub + mul + v_exp_f32 with free
// source-neg modifier = 2 VALU + 1 TRANS per element (+1/2 v_pk_add_f32).
// One v_wmma_f32_16x16x4_f32 against an all-ones B reduces the per-lane
// partials into per-row sums at the end.
// ---------------------------------------------------------------------------
__global__ __launch_bounds__(256)
void gmm_kde_kernel(const float* __restrict__ x, float* __restrict__ S, float kscale)
{
    __shared__ float sx[ISLICE];
    const int tid = threadIdx.x;
    const int i0  = blockIdx.y * ISLICE;
    for (int idx = tid; idx < ISLICE; idx += 256) sx[idx] = x[i0 + idx] * kscale;
    __syncthreads();

    const int lane     = tid & 31;
    const int wave     = tid >> 5;
    const int row_base = blockIdx.x * ROWS_PER_BLOCK + wave * 16;
    const int h        = lane >> 4;                 // A-matrix K-pair selector
    const float xjk    = x[row_base + (lane & 15)] * kscale;

    float a0 = 0.0f, a1 = 0.0f, a2 = 0.0f, a3 = 0.0f;   // 4 independent chains
    const float4* sx4 = (const float4*)sx;
    #pragma unroll 4
    for (int t = 0; t < ISLICE / 4; t += 2) {
        const float4 xi = sx4[t + h];               // 2-way LDS broadcast read
        const float r0 = xjk - xi.x, r1 = xjk - xi.y, r2 = xjk - xi.z, r3 = xjk - xi.w;
        a0 += __builtin_amdgcn_exp2f(-(r0 * r0));   // neg folds into v_exp src mod
        a1 += __builtin_amdgcn_exp2f(-(r1 * r1));
        a2 += __builtin_amdgcn_exp2f(-(r2 * r2));
        a3 += __builtin_amdgcn_exp2f(-(r3 * r3));
    }

    // A[row][2h + {0,1}] = partial sums; B = ones -> D[m][n] = row sum of m
    v2f A; A[0] = a0 + a2; A[1] = a1 + a3;
    v2f B; B[0] = 1.0f;    B[1] = 1.0f;
    v8f C = {};
    v8f D = __builtin_amdgcn_wmma_f32_16x16x4_f32(
        /*neg_a=*/false, A, /*neg_b=*/false, B, /*c_mod=*/(short)0, C,
        /*reuse_a=*/false, /*reuse_b=*/false);

    // D layout: lane n<16 holds D[M=v][N=n] in component v; lane n+16 holds M=v+8.
    // Lane 0 therefore has rows 0..7, lane 16 rows 8..15 (any N column works).
    if (lane == 0) {
        #pragma unroll
        for (int v = 0; v < 8; ++v) atomicAdd(&S[row_base + v], D[v]);
    } else if (lane == 16) {
        #pragma unroll
        for (int v = 0; v < 8; ++v) atomicAdd(&S[row_base + 8 + v], D[v]);
    }
}

// ---------------------------------------------------------------------------
// Kernel 3: mixture pdf per point, squared error vs KDE pdf, reduce to scalar
// ---------------------------------------------------------------------------
__global__ __launch_bounds__(256)
void gmm_final_kernel(const float* __restrict__ x,
                      const float* __restrict__ means,
                      const float* __restrict__ coef,
                      const float* __restrict__ e2k,
                      const float* __restrict__ S,
                      float* __restrict__ out, float inv_norm)
{
    __shared__ float sc[NCOMP], se[NCOMP], sm[NCOMP];
    const int tid = threadIdx.x;
    if (tid < NCOMP) { sc[tid] = coef[tid]; se[tid] = e2k[tid]; sm[tid] = means[tid]; }
    __syncthreads();

    const int j = blockIdx.x * 256 + tid;
    const float xj = x[j];
    float mix = 0.0f;
    #pragma unroll 8
    for (int k = 0; k < NCOMP; ++k) {
        const float d = xj - sm[k];
        mix += sc[k] * __builtin_amdgcn_exp2f(d * d * se[k]);
    }
    const float diff = mix - S[j] * inv_norm;
    float sq = diff * diff;

    __shared__ float red[256];
    red[tid] = sq; __syncthreads();
    for (int s = 128; s > 0; s >>= 1) {
        if (tid < s) red[tid] += red[tid + s];
        __syncthreads();
    }
    if (tid == 0) atomicAdd(out, red[0]);
}

// ---------------------------------------------------------------------------
extern "C" void kernel_launch(void* const* d_in, const int* in_sizes, int n_in,
                              void* d_out, int out_size, void* d_ws, size_t ws_size,
                              hipStream_t stream)
{
    const float* x  = (const float*)d_in[0];   // [8192]
    const float* wl = (const float*)d_in[1];   // [64] weight logits
    const float* mu = (const float*)d_in[2];   // [64] means
    const float* lv = (const float*)d_in[3];   // [64] log vars
    float* out = (float*)d_out;                // scalar f32

    float* coef = (float*)d_ws;                // [64]
    float* e2k  = coef + NCOMP;                // [64]
    float* S    = e2k + NCOMP;                 // [8192] KDE row sums

    // bandwidth = 0.5*128/N = 0.0078125 ; -0.5/bw^2 = -8192
    const double bw   = 0.5 * 128.0 / (double)NPTS;
    const double bw2  = bw * bw;
    const double c2   = -0.5 / bw2 * 1.4426950408889634;                       // exp2 coeff (negative)
    const float  kscale   = (float)sqrt(-c2);                                  // sqrt(8192*log2e)
    const float  inv_norm = (float)(1.0 / (sqrt(6.283185307179586 * bw2) * (double)NPTS));

    gmm_prep_kernel<<<1, NCOMP, 0, stream>>>(wl, lv, coef, e2k, S, out);
    gmm_kde_kernel<<<dim3(NPTS / ROWS_PER_BLOCK, ISPLIT), 256, 0, stream>>>(x, S, kscale);
    gmm_final_kernel<<<NPTS / 256, 256, 0, stream>>>(x, mu, coef, e2k, S, out, inv_norm);
}